// SinhLinearAttention_79164837200631
// MI455X (gfx1250) — compile-verified
//
#include <hip/hip_runtime.h>
#include <hip/hip_bf16.h>

// SinhLinearAttention on MI455X (gfx1250), fp32 end-to-end.
//
// Roofline: 4.3 GFLOP vs 268 MB of HBM traffic (AI ~16 FLOP/B) -> memory
// bound at 23.3 TB/s (~11.5us floor). V_WMMA_F32_16X16X4_F32 keeps reference
// precision; both einsums run on the f32 WMMA path. Phase-1 V traffic uses the
// CDNA5 async global->LDS path (global_load_async_to_lds_b128 / ASYNCcnt) with
// double-buffered LDS so HBM latency hides under featurize + WMMA work.
// The [64x64]+[64] per-head intermediates (~1 MB) stay resident in L2 (192 MB).

typedef float v2f __attribute__((ext_vector_type(2)));
typedef float v8f __attribute__((ext_vector_type(8)));

#define B_ 4
#define H_ 16
#define S_ 4096
#define D_ 64
#define SPLIT 8                    // split-K blocks per (b,h) in phase 1
#define SCHUNK (S_ / SPLIT)        // 512 rows per split
#define PITCH 72                   // LDS row pitch in words; 72%64=8 staggers
                                   // banks between the two half-wave row groups
#define WS_STRIDE (D_ * D_ + D_)   // per-head ws: M[64x64] then ksum[64]

__device__ __forceinline__ float relu_sinh(float x) {
  // sinh is odd & monotonic -> relu(sinh(x)) = (x>0) ? sinh(x) : 0
  return x > 0.f ? 0.5f * (__expf(x) - __expf(-x)) : 0.f;
}

// D = A(16x4 f32) * B(4x16 f32) + C(16x16 f32), wave32.
__device__ __forceinline__ v8f wmma_f32(v2f a, v2f b, v8f c) {
  return __builtin_amdgcn_wmma_f32_16x16x4_f32(
      /*neg_a=*/false, a, /*neg_b=*/false, b,
      /*c_mod=*/(short)0, c, /*reuse_a=*/false, /*reuse_b=*/false);
}

// CDNA5 async copy: 16B global -> LDS per lane, tracked by ASYNCcnt.
// LDS generic address low 32 bits == LDS offset (ISA 10.2 aperture rules).
__device__ __forceinline__ void async_copy_b128(unsigned lds_addr,
                                                const float* gaddr) {
  asm volatile("global_load_async_to_lds_b128 %0, %1, off"
               :: "v"(lds_addr), "v"(gaddr)
               : "memory");
}
#define WAIT_ASYNCCNT(n) asm volatile("s_wait_asynccnt " #n ::: "memory")

__global__ __launch_bounds__(256) void zero_ws_kernel(float* ws, int n) {
  int i = blockIdx.x * 256 + threadIdx.x;
  if (i < n) ws[i] = 0.f;
}

// Phase 1: M[bh] += Kf_chunk^T * V_chunk ; ksum[bh] += sum_s Kf_chunk.
// grid = B*H*SPLIT blocks of 256 threads (8 waves).
// Wave w owns C tiles (m_tile = w&3, n_tiles = (w>>2)*2 + {0,1}).
// Software pipeline: V(i+1) via async->LDS, K(i+1)/mask(i+1) staged in regs,
// both issued before the WMMA work on chunk i.
__global__ __launch_bounds__(256) void p1_kTv(const float* __restrict__ Kp,
                                              const float* __restrict__ Vp,
                                              const unsigned char* __restrict__ maskp,
                                              float* __restrict__ ws) {
  __shared__ __align__(16) float kf[2][16 * PITCH];   // featurized K chunks
  __shared__ __align__(16) float vt[2][16 * PITCH];   // V chunks (async-filled)
  __shared__ __align__(16) float kpart[16 * 64];      // ksum partials

  const int bh    = blockIdx.x / SPLIT;
  const int chunk = blockIdx.x % SPLIT;
  const int b     = bh / H_;
  const int t     = threadIdx.x;
  const int wave  = t >> 5;
  const int lane  = t & 31;
  const int m_tile = wave & 3;
  const int n_tile = (wave >> 2) * 2;

  const int row = t >> 4;            // 0..15: chunk row this thread loads
  const int cg  = (t & 15) << 2;     // col group (float4)

  const long base = (long)bh * S_ * D_ + (long)(chunk * SCHUNK) * D_;
  const float* Kg = Kp + base + (long)row * D_ + cg;
  const float* Vg = Vp + base + (long)row * D_ + cg;
  const unsigned char* mrow = maskp + (long)b * S_ + chunk * SCHUNK + row;

  const unsigned ldsV0 = (unsigned)(unsigned long long)&vt[0][row * PITCH + cg];
  const unsigned ldsV1 = (unsigned)(unsigned long long)&vt[1][row * PITCH + cg];

  v8f c0 = {}; v8f c1 = {};
  float ks0 = 0.f, ks1 = 0.f, ks2 = 0.f, ks3 = 0.f;

  const int ml = lane & 15;
  const int rA = (lane >> 4) << 1;   // 0 for lanes 0-15, 2 for lanes 16-31

  // ---- prologue: chunk 0 in flight (V -> LDS async, K/mask -> regs)
  async_copy_b128(ldsV0, Vg);
  float4 kreg = *(const float4*)Kg;
  unsigned char mreg = *mrow;

  int p = 0;
  for (int s0 = 0; s0 < SCHUNK; s0 += 16, p ^= 1) {
    // ---- featurize staged K chunk into kf[p]
    const float mv = (float)mreg;
    const float f0 = relu_sinh(kreg.x) * mv;
    const float f1 = relu_sinh(kreg.y) * mv;
    const float f2 = relu_sinh(kreg.z) * mv;
    const float f3 = relu_sinh(kreg.w) * mv;
    *(float4*)(&kf[p][row * PITCH + cg]) = make_float4(f0, f1, f2, f3);
    ks0 += f0; ks1 += f1; ks2 += f2; ks3 += f3;

    // ---- issue chunk i+1 (overlaps with barrier + WMMA below), then make
    //      sure chunk i's async V landed (in-order completion => cnt<=1).
    if (s0 + 16 < SCHUNK) {
      async_copy_b128(p ? ldsV0 : ldsV1, Vg + (long)(s0 + 16) * D_);
      kreg = *(const float4*)(Kg + (long)(s0 + 16) * D_);
      mreg = mrow[s0 + 16];
      WAIT_ASYNCCNT(0x1);
    } else {
      WAIT_ASYNCCNT(0x0);
    }
    __syncthreads();

    // ---- 4 WMMA K-steps over the 16 staged rows of buffer p
#pragma unroll
    for (int kk = 0; kk < 4; ++kk) {
      const int r0 = kk * 4 + rA;
      v2f a, b0, b1;
      // A = Kf^T tile: A[m][k] = kf[s=k][d=m_base+m]
      a.x  = kf[p][(r0 + 0) * PITCH + m_tile * 16 + ml];
      a.y  = kf[p][(r0 + 1) * PITCH + m_tile * 16 + ml];
      // B = V tile: B[k][n] = vt[s=k][e=n_base+n]
      b0.x = vt[p][(r0 + 0) * PITCH + n_tile * 16 + ml];
      b0.y = vt[p][(r0 + 1) * PITCH + n_tile * 16 + ml];
      b1.x = vt[p][(r0 + 0) * PITCH + (n_tile + 1) * 16 + ml];
      b1.y = vt[p][(r0 + 1) * PITCH + (n_tile + 1) * 16 + ml];
      c0 = wmma_f32(a, b0, c0);
      c1 = wmma_f32(a, b1, c1);
    }
    // second barrier: next iteration's ds_store/async writes to buffer p^1
    // must not race other waves' WMMA reads still in flight this iteration.
    __syncthreads();
  }

  // ---- ksum: reduce the 16 row-partials per column, atomically add
  *(float4*)(kpart + row * 64 + cg) = make_float4(ks0, ks1, ks2, ks3);
  __syncthreads();
  float* M = ws + (long)bh * WS_STRIDE;
  if (t < 64) {
    float s = 0.f;
#pragma unroll
    for (int r = 0; r < 16; ++r) s += kpart[r * 64 + t];
    atomicAdd(M + D_ * D_ + t, s);
  }

  // ---- flush C tiles (split-K accumulate across blocks)
  const int mrow0 = m_tile * 16 + ((lane >= 16) ? 8 : 0);
  const int ncol  = n_tile * 16 + ml;
#pragma unroll
  for (int v = 0; v < 8; ++v) {
    atomicAdd(M + (mrow0 + v) * D_ + ncol,      c0[v]);
    atomicAdd(M + (mrow0 + v) * D_ + ncol + 16, c1[v]);
  }
}

// Phase 2: out = (Qf @ M) / max(Qf . ksum, eps).
// grid = B*H*(S/128) blocks; each wave produces a 16x64 output tile.
#define ROWS_PER_BLOCK 128
__global__ __launch_bounds__(256) void p2_qM(const float* __restrict__ Qp,
                                             const float* __restrict__ ws,
                                             float* __restrict__ out) {
  __shared__ __align__(16) float Ms[64 * PITCH];        // K^T V  (18.4 KB)
  __shared__ float ksums[64];
  __shared__ __align__(16) float qf[8][16 * PITCH];     // per-wave Qf tiles
  __shared__ float norms[8][16];

  const int nSB = S_ / ROWS_PER_BLOCK;
  const int bh  = blockIdx.x / nSB;
  const int sb  = blockIdx.x % nSB;
  const int t = threadIdx.x, wave = t >> 5, lane = t & 31;
  const int ml = lane & 15;
  const int rA = (lane >> 4) << 1;

  // ---- stage M (64x64) + ksum into LDS (reads hit L2 from phase 1)
  const float* M = ws + (long)bh * WS_STRIDE;
  for (int i = t; i < 64 * 16; i += 256) {
    const int r = i >> 4, c = (i & 15) << 2;
    *(float4*)(Ms + r * PITCH + c) = *(const float4*)(M + r * 64 + c);
  }
  if (t < 64) ksums[t] = M[D_ * D_ + t];

  // ---- featurize this wave's 16x64 Q tile into LDS
  const int r_base = sb * ROWS_PER_BLOCK + wave * 16;
  const float* Qb = Qp + (long)bh * S_ * D_;
  for (int i = lane; i < 16 * 16; i += 32) {
    const int r = i >> 4, c = (i & 15) << 2;
    float4 q = *(const float4*)(Qb + (long)(r_base + r) * D_ + c);
    q.x = relu_sinh(q.x); q.y = relu_sinh(q.y);
    q.z = relu_sinh(q.z); q.w = relu_sinh(q.w);
    *(float4*)(&qf[wave][r * PITCH + c]) = q;
  }
  __syncthreads();

  // ---- normalizer: norm[row] = max(Qf[row] . ksum, eps)
  if (lane < 16) {
    float acc = 0.f;
#pragma unroll 8
    for (int d = 0; d < 64; ++d) acc += qf[wave][lane * PITCH + d] * ksums[d];
    norms[wave][lane] = fmaxf(acc, 1e-6f);
  }
  __syncthreads();

  // ---- 16x64 = [16x64]x[64x64] via 16 K-steps x 4 N-tiles of WMMA f32
  v8f acc[4] = {};
#pragma unroll 4
  for (int kk = 0; kk < 16; ++kk) {
    const int k0 = kk * 4 + rA;
    v2f a;
    a.x = qf[wave][ml * PITCH + k0 + 0];
    a.y = qf[wave][ml * PITCH + k0 + 1];
#pragma unroll
    for (int nt = 0; nt < 4; ++nt) {
      v2f bb;
      bb.x = Ms[(k0 + 0) * PITCH + nt * 16 + ml];
      bb.y = Ms[(k0 + 1) * PITCH + nt * 16 + ml];
      acc[nt] = wmma_f32(a, bb, acc[nt]);
    }
  }

  // ---- scale by 1/norm and store (coalesced 16-lane runs)
  float* ob = out + (long)bh * S_ * D_;
  const int half = (lane >= 16) ? 8 : 0;
#pragma unroll
  for (int v = 0; v < 8; ++v) {
    const int r = v + half;
    const float inv = 1.0f / norms[wave][r];
    const long ro = (long)(r_base + r) * D_;
#pragma unroll
    for (int nt = 0; nt < 4; ++nt)
      ob[ro + nt * 16 + ml] = acc[nt][v] * inv;
  }
}

extern "C" void kernel_launch(void* const* d_in, const int* in_sizes, int n_in,
                              void* d_out, int out_size, void* d_ws, size_t ws_size,
                              hipStream_t stream) {
  (void)in_sizes; (void)n_in; (void)out_size; (void)ws_size;
  const float* Q = (const float*)d_in[0];
  const float* K = (const float*)d_in[1];
  const float* V = (const float*)d_in[2];
  const unsigned char* mask = (const unsigned char*)d_in[3];  // jnp bool -> 1B
  float* out = (float*)d_out;
  float* ws  = (float*)d_ws;

  const int nWS = B_ * H_ * WS_STRIDE;  // 266,240 floats (~1.04 MB)
  zero_ws_kernel<<<(nWS + 255) / 256, 256, 0, stream>>>(ws, nWS);
  p1_kTv<<<B_ * H_ * SPLIT, 256, 0, stream>>>(K, V, mask, ws);
  p2_qM<<<B_ * H_ * (S_ / ROWS_PER_BLOCK), 256, 0, stream>>>(Q, ws, out);
}